// CRF_74809740362319
// MI455X (gfx1250) — compile-verified
//
#include <hip/hip_runtime.h>

typedef __attribute__((ext_vector_type(2))) float v2f;
typedef __attribute__((ext_vector_type(4))) float v4f;
typedef __attribute__((ext_vector_type(8))) float v8f;

#define BB   512
#define TT   1024
#define KK   48
#define TMIN 512   // lengths >= T/2, so mask[b][t] is guaranteed true for t < 512

#define INV_LN2 1.44269504088896340736f
#define LN2     0.69314718055994530942f

// --------------------------------------------------------------------------
// Kernel A: numerator path score per batch -> ws[b].
// One wave per batch; also pulls emission cachelines into the 192MB L2
// before kernel B streams them.
// --------------------------------------------------------------------------
__global__ __launch_bounds__(128) void crf_num(
    const float* __restrict__ em, const int* __restrict__ tags,
    const unsigned char* __restrict__ mask, const float* __restrict__ trans,
    const float* __restrict__ st, const float* __restrict__ en,
    float* __restrict__ ws)
{
  const int lane = threadIdx.x & 31;
  const int b    = (blockIdx.x * blockDim.x + threadIdx.x) >> 5;
  if (b >= BB) return;
  const int* tg = tags + b * TT;
  const unsigned char* mk = mask + (size_t)b * TT;

  float s = 0.f;
  int cnt = 0;
  for (int t = lane; t < TT; t += 32) {
    int tag = tg[t];
    int m = (t < TMIN) ? 1 : (int)mk[t];
    cnt += m;
    float e = em[((size_t)b * TT + t) * KK + tag];
    if (t == 0) {
      s += e;                                  // mask[0] always true
    } else {
      float tr = trans[tg[t - 1] * KK + tag];  // trans[prev, next]
      s += m ? (e + tr) : 0.f;
    }
  }
  #pragma unroll
  for (int off = 16; off > 0; off >>= 1) {
    s   += __shfl_xor(s, off, 32);
    cnt += __shfl_xor(cnt, off, 32);
  }
  if (lane == 0) {
    s += st[tg[0]] + en[tg[cnt - 1]];
    ws[b] = s;
  }
}

// --------------------------------------------------------------------------
// Kernel B: forward algorithm (log denominator), 16 chains per wave, log2 domain.
//
//   nxt[b,j] = log2( sum_i 2^(al2[b,i]-mx2) * exp(trans[j,i]) ) + mx2 + e[b,j]/ln2
//   nxt_T (48x16) = E (48x48) @ P_T (48x16)   via V_WMMA_F32_16X16X4_F32
//
// State labeling trick: B operands are a pure register re-grouping of the
// C-layout alpha registers (no cross-lane movement). The induced state
// permutation on the WMMA K axis,
//     kstate(c,w,half) = 16*((2c+w)/8) + (2c+w)%8 + 8*half,
// is baked into the constant A operand gather, so per-step reshuffling
// costs zero instructions.
// --------------------------------------------------------------------------
__global__ __launch_bounds__(32) void crf_fwd(
    const float* __restrict__ em, const unsigned char* __restrict__ mask,
    const float* __restrict__ trans, const float* __restrict__ st,
    const float* __restrict__ en, const float* __restrict__ ws,
    float* __restrict__ out)
{
  const int lane = threadIdx.x & 31;
  const int half = lane >> 4;
  const int n    = lane & 15;        // batch within tile == WMMA N == lane%16
  const int b0   = blockIdx.x * 16;
  const int b    = b0 + n;

  // ---- constant A operands: E[j, kstate] = exp(trans[j, kstate]), A-layout ----
  v2f A[3][12];
  #pragma unroll
  for (int tau = 0; tau < 3; ++tau)
    #pragma unroll
    for (int c = 0; c < 12; ++c)
      #pragma unroll
      for (int w = 0; w < 2; ++w) {
        const int j  = 2 * c + w;                         // flat P-register slot
        const int ks = (j / 8) * 16 + (j % 8) + 8 * half; // permuted input state
        A[tau][c][w] = __expf(trans[(tau * 16 + n) * KK + ks]);
      }

  // ---- alpha (log2 domain), C-layout: al[tau][v] = state tau*16+v+8*half ----
  const float* emrow0 = em + (size_t)b * TT * KK;
  float al[3][8];
  #pragma unroll
  for (int tau = 0; tau < 3; ++tau) {
    const int sb = tau * 16 + 8 * half;
    v4f x0 = *(const v4f*)(emrow0 + sb);
    v4f x1 = *(const v4f*)(emrow0 + sb + 4);
    #pragma unroll
    for (int v = 0; v < 4; ++v) {
      al[tau][v]     = (st[sb + v]     + x0[v]) * INV_LN2;
      al[tau][v + 4] = (st[sb + 4 + v] + x1[v]) * INV_LN2;
    }
  }

  // per-chain max (batch == lane%16, so only one cross-half shuffle)
  auto maxAll = [&]() {
    float q[24];
    #pragma unroll
    for (int i = 0; i < 24; ++i) q[i] = al[i / 8][i % 8];
    #pragma unroll
    for (int i = 0; i < 12; ++i) q[i] = fmaxf(q[i], q[i + 12]);
    #pragma unroll
    for (int i = 0; i < 6; ++i)  q[i] = fmaxf(q[i], q[i + 6]);
    #pragma unroll
    for (int i = 0; i < 3; ++i)  q[i] = fmaxf(q[i], q[i + 3]);
    float mx = fmaxf(fmaxf(q[0], q[1]), q[2]);
    return fmaxf(mx, __shfl_xor(mx, 16, 32));
  };

  // emissions for one step, C-layout halves: 6 contiguous b128 loads
  auto loadE = [&](v4f (&dst)[3][2], int t) {
    const float* p = em + ((size_t)b * TT + t) * KK + 8 * half;
    #pragma unroll
    for (int tau = 0; tau < 3; ++tau) {
      dst[tau][0] = *(const v4f*)(p + tau * 16);
      dst[tau][1] = *(const v4f*)(p + tau * 16 + 4);
    }
  };

  float mx2 = 0.f;   // per-chain running max (log2 domain), refreshed every 4 steps

  auto step = [&](int t, v4f (&ecur)[3][2], v4f (&enext)[3][2], int tnext,
                  bool use_mask, bool refresh) {
    loadE(enext, tnext);                       // prefetch next step first
    int mv = 1;
    if (use_mask) mv = (int)mask[(size_t)b * TT + t];
    if (refresh)  mx2 = maxAll();

    // B operands: P = 2^(al - mx2); pure register re-grouping, no shuffles
    v2f Bop[12];
    #pragma unroll
    for (int c = 0; c < 12; ++c)
      #pragma unroll
      for (int w = 0; w < 2; ++w) {
        const int j = 2 * c + w;
        Bop[c][w] = __builtin_amdgcn_exp2f(al[j / 8][j % 8] - mx2);
      }

    // nxt_T = E @ P : 3 accumulator tiles x 12 K-chunks of f32 WMMA
    v8f acc0 = {}, acc1 = {}, acc2 = {};
    #pragma unroll
    for (int c = 0; c < 12; ++c) {
      acc0 = __builtin_amdgcn_wmma_f32_16x16x4_f32(false, A[0][c], false, Bop[c],
                                                   (short)0, acc0, false, false);
      acc1 = __builtin_amdgcn_wmma_f32_16x16x4_f32(false, A[1][c], false, Bop[c],
                                                   (short)0, acc1, false, false);
      acc2 = __builtin_amdgcn_wmma_f32_16x16x4_f32(false, A[2][c], false, Bop[c],
                                                   (short)0, acc2, false, false);
    }

    // alpha = mask ? log2(acc) + mx2 + e_t/ln2 : alpha
    #pragma unroll
    for (int v = 0; v < 8; ++v) {
      float n0 = fmaf(ecur[0][v / 4][v % 4], INV_LN2,
                      __builtin_amdgcn_logf(acc0[v]) + mx2);
      float n1 = fmaf(ecur[1][v / 4][v % 4], INV_LN2,
                      __builtin_amdgcn_logf(acc1[v]) + mx2);
      float n2 = fmaf(ecur[2][v / 4][v % 4], INV_LN2,
                      __builtin_amdgcn_logf(acc2[v]) + mx2);
      if (use_mask) {
        al[0][v] = mv ? n0 : al[0][v];
        al[1][v] = mv ? n1 : al[1][v];
        al[2][v] = mv ? n2 : al[2][v];
      } else {
        al[0][v] = n0;
        al[1][v] = n1;
        al[2][v] = n2;
      }
    }
  };

  // ---- main scan, software-pipelined, unrolled by 4 (max refresh period) ----
  v4f ea[3][2], eb[3][2];
  loadE(ea, 1);

  // phase 1: t in [1, 512), mask guaranteed true -> no mask loads/selects
  int t = 1;
  #pragma unroll 1
  for (; t + 3 < TMIN; t += 4) {
    step(t,     ea, eb, t + 1, false, true);
    step(t + 1, eb, ea, t + 2, false, false);
    step(t + 2, ea, eb, t + 3, false, false);
    step(t + 3, eb, ea, t + 4, false, false);
  }
  // tail: t = 509, 510, 511
  step(t,     ea, eb, t + 1, false, true);
  step(t + 1, eb, ea, t + 2, false, false);
  step(t + 2, ea, eb, t + 3, false, false);   // loads e_512 into eb
  t += 3;                                     // t == 512, current buffer: eb

  // phase 2: t in [512, 1024), masked; 512 steps = 128 groups of 4
  #pragma unroll 1
  for (; t + 3 < TT; t += 4) {
    step(t,     eb, ea, t + 1, true, true);
    step(t + 1, ea, eb, t + 2, true, false);
    step(t + 2, eb, ea, t + 3, true, false);
    step(t + 3, ea, eb, (t + 4 < TT) ? (t + 4) : (TT - 1), true, false);
  }

  // ---- log_den = logsumexp(alpha + end) per chain ----
  #pragma unroll
  for (int tau = 0; tau < 3; ++tau) {
    const int sb = tau * 16 + 8 * half;
    #pragma unroll
    for (int v = 0; v < 8; ++v)
      al[tau][v] = fmaf(en[sb + v], INV_LN2, al[tau][v]);
  }
  float mxf = maxAll();
  float ssum = 0.f;
  #pragma unroll
  for (int tau = 0; tau < 3; ++tau)
    #pragma unroll
    for (int v = 0; v < 8; ++v)
      ssum += __builtin_amdgcn_exp2f(al[tau][v] - mxf);
  ssum += __shfl_xor(ssum, 16, 32);
  float logden = (__builtin_amdgcn_logf(ssum) + mxf) * LN2;

  if (half == 0) out[b] = ws[b] - logden;
}

// --------------------------------------------------------------------------
extern "C" void kernel_launch(void* const* d_in, const int* in_sizes, int n_in,
                              void* d_out, int out_size, void* d_ws, size_t ws_size,
                              hipStream_t stream) {
  const float*         em    = (const float*)d_in[0];          // (512,1024,48) f32
  const int*           tags  = (const int*)d_in[1];            // (512,1024) i32
  const unsigned char* mask  = (const unsigned char*)d_in[2];  // (512,1024) bool (1B)
  const float*         trans = (const float*)d_in[3];          // (48,48)
  const float*         st    = (const float*)d_in[4];          // (48,)
  const float*         en    = (const float*)d_in[5];          // (48,)
  float* ws  = (float*)d_ws;    // numerator scores, 512 floats
  float* out = (float*)d_out;   // 512 floats

  crf_num<<<128, 128, 0, stream>>>(em, tags, mask, trans, st, en, ws);
  crf_fwd<<<BB / 16, 32, 0, stream>>>(em, mask, trans, st, en, ws, out);
}